// FocalCTCLoss_19499151524231
// MI455X (gfx1250) — compile-verified
//
#include <hip/hip_runtime.h>
#include <stdint.h>

// Problem constants (match reference shapes exactly)
#define T_DIM 2000
#define B_DIM 64
#define C_DIM 256
#define S_DIM 100
#define E_DIM 201           // 2*S + 1
#define NEGV  (-1e30f)
#define NBUF  4             // async LDS ring depth (>=3 needed for hazard safety)

#define LDS_AS __attribute__((address_space(3)))
#define GA_AS  __attribute__((address_space(1)))

// ---------------------------------------------------------------------------
// CDNA5 async global->LDS copy + asynccnt waits.
// Probe round 1 confirmed __builtin_amdgcn_global_load_async_to_lds_b32 exists
// with 4 args and takes (AS1 int* gsrc, AS3 int* ldst, int offset, int cpol)
// (clang prints the AS1 param as "__device__ int *"). Pointers are built via
// integer casts:
//   - generic shared ptr low 32 bits == LDS byte offset (aperture spec)
//   - generic global ptr == 64-bit VA == AS1 VA
// ---------------------------------------------------------------------------
#if defined(__HIP_DEVICE_COMPILE__)

#if __has_builtin(__builtin_amdgcn_global_load_async_to_lds_b32)
#define ASYNC_LOAD_F32(gp, lp_)                                                \
  __builtin_amdgcn_global_load_async_to_lds_b32(                               \
      (GA_AS int*)(uintptr_t)(gp),                                             \
      (LDS_AS int*)(uint32_t)(uintptr_t)(lp_), 0, 0)
#else
#define ASYNC_LOAD_F32(gp, lp_)                                                \
  asm volatile("global_load_async_to_lds_b32 %0, %1, off"                      \
               :: "v"((uint32_t)(uintptr_t)(lp_)),                             \
                  "v"((uint64_t)(uintptr_t)(gp))                               \
               : "memory")
#endif

#if __has_builtin(__builtin_amdgcn_s_wait_asynccnt)
#define WAIT_ASYNC_LE1() __builtin_amdgcn_s_wait_asynccnt(1)
#define WAIT_ASYNC_LE0() __builtin_amdgcn_s_wait_asynccnt(0)
#else
#define WAIT_ASYNC_LE1() asm volatile("s_wait_asynccnt 1" ::: "memory")
#define WAIT_ASYNC_LE0() asm volatile("s_wait_asynccnt 0" ::: "memory")
#endif

#else  // host pass: no-ops so the host semantic pass compiles
#define ASYNC_LOAD_F32(gp, lp_) ((void)0)
#define WAIT_ASYNC_LE1()        ((void)0)
#define WAIT_ASYNC_LE0()        ((void)0)
#endif

// ---------------------------------------------------------------------------
// Kernel 1: one workgroup per batch sample b; thread s owns DP state s.
// alpha double-buffered in LDS with 2-entry NEG pad (branch-free s-1/s-2).
// log_probs rows streamed into a 4-deep LDS ring via async loads.
// One barrier per timestep.
// ---------------------------------------------------------------------------
__global__ __launch_bounds__(256) void FocalCTC_dp_kernel(
    const float* __restrict__ log_probs,   // (T, B, C)
    const int*   __restrict__ targets,     // (B, S)
    const int*   __restrict__ input_len,   // (B,)
    const int*   __restrict__ target_len,  // (B,)
    float*       __restrict__ per_sample)  // (B,) focal loss per sample
{
  __shared__ float lp[NBUF][C_DIM];        // staged log-prob rows
  __shared__ float Abuf[2][E_DIM + 2];     // alpha, +2 NEG pad at front
  __shared__ int   eidx[E_DIM];            // ext label class per state
  __shared__ int   eskip[E_DIM];           // skip-transition allowed per state

  const int b   = blockIdx.x;
  const int tid = threadIdx.x;
  const int len = input_len[b];            // in [T/2, T], >= 1
  const int tl  = target_len[b];           // in [S/2, S]
  const float* base = log_probs + (size_t)b * C_DIM;  // row t at +t*B*C

  // Extended-label table: even s -> blank(0); odd s -> targets[b][s>>1].
  // skip[s] = s odd, s>=3, targets differ (targets are never blank).
  if (tid < E_DIM) {
    int id = 0, sk = 0;
    if (tid & 1) {
      const int j = tid >> 1;
      id = targets[b * S_DIM + j];
      if (tid >= 3) sk = (id != targets[b * S_DIM + j - 1]) ? 1 : 0;
    }
    eidx[tid]  = id;
    eskip[tid] = sk;
  }
  if (tid < 2) { Abuf[0][tid] = NEGV; Abuf[1][tid] = NEGV; }  // pads

  // Prologue: async-stage rows 0 (and 1), guarantee row 0 landed.
  ASYNC_LOAD_F32(base + /*t=0*/ (size_t)0 + tid, &lp[0][tid]);
  if (len > 1) {
    ASYNC_LOAD_F32(base + (size_t)1 * B_DIM * C_DIM + tid, &lp[1][tid]);
    WAIT_ASYNC_LE1();
  } else {
    WAIT_ASYNC_LE0();
  }
  __syncthreads();

  // t = 0 init: alpha0[0] = lp0[blank], alpha0[1] = lp0[label0], else NEG.
  if (tid < E_DIM) {
    Abuf[0][tid + 2] = (tid <= 1) ? lp[0][eidx[tid]] : NEGV;
  }

  int cur = 0;
  for (int t = 1; t < len; ++t) {          // t >= len never changes alpha
    if (t + 1 < len) {
      ASYNC_LOAD_F32(base + (size_t)(t + 1) * B_DIM * C_DIM + tid,
                     &lp[(t + 1) % NBUF][tid]);
      WAIT_ASYNC_LE1();                    // in-order per wave => row t landed
    } else {
      WAIT_ASYNC_LE0();
    }
    __syncthreads();                       // alpha(t-1) + row t visible to all

    if (tid < E_DIM) {
      const float e  = lp[t % NBUF][eidx[tid]];
      const float a0 = Abuf[cur][tid + 2];
      const float a1 = Abuf[cur][tid + 1];
      const float a2 = eskip[tid] ? Abuf[cur][tid] : NEGV;
      const float m  = fmaxf(a0, fmaxf(a1, a2));
      const float r  = m + logf(expf(a0 - m) + expf(a1 - m) + expf(a2 - m));
      Abuf[cur ^ 1][tid + 2] = r + e;
    }
    cur ^= 1;
  }
  __syncthreads();

  // fin is captured at t = len-1, which is exactly the final alpha buffer.
  if (tid == 0) {
    const int   i1 = 2 * tl - 1;           // last label state
    const int   i2 = 2 * tl;               // trailing blank state
    const float a  = Abuf[cur][i1 + 2];
    const float c  = Abuf[cur][i2 + 2];
    const float m  = fmaxf(a, c);
    const float fin = m + logf(expf(a - m) + expf(c - m));

    float l = (fin < NEGV * 0.5f) ? 0.0f : -fin;   // zero_infinity
    l = fmaxf(l, 1e-6f);                            // clamp(min=1e-6)
    float p = expf(-l);
    p = fminf(fmaxf(p, 1e-6f), 1.0f);               // clip
    const float om = 1.0f - p;
    per_sample[b] = l * (om * om);                  // ALPHA=1, GAMMA=2
  }
}

// ---------------------------------------------------------------------------
// Kernel 2: deterministic fixed-tree mean over the 64 per-sample losses.
// ---------------------------------------------------------------------------
__global__ __launch_bounds__(64) void FocalCTC_mean_kernel(
    const float* __restrict__ per_sample, float* __restrict__ out)
{
  __shared__ float sm[B_DIM];
  const int tid = threadIdx.x;
  sm[tid] = per_sample[tid];
  __syncthreads();
  for (int ofs = B_DIM / 2; ofs > 0; ofs >>= 1) {
    if (tid < ofs) sm[tid] += sm[tid + ofs];
    __syncthreads();
  }
  if (tid == 0) out[0] = sm[0] * (1.0f / (float)B_DIM);
}

// ---------------------------------------------------------------------------
extern "C" void kernel_launch(void* const* d_in, const int* in_sizes, int n_in,
                              void* d_out, int out_size, void* d_ws, size_t ws_size,
                              hipStream_t stream) {
  (void)in_sizes; (void)n_in; (void)out_size; (void)ws_size;
  const float* log_probs = (const float*)d_in[0];
  const int*   targets   = (const int*)d_in[1];
  const int*   in_len    = (const int*)d_in[2];
  const int*   tgt_len   = (const int*)d_in[3];
  float* per_sample = (float*)d_ws;        // 64 floats of scratch
  float* out        = (float*)d_out;

  FocalCTC_dp_kernel<<<B_DIM, 256, 0, stream>>>(log_probs, targets, in_len,
                                                tgt_len, per_sample);
  FocalCTC_mean_kernel<<<1, B_DIM, 0, stream>>>(per_sample, out);
}